// GCNResidual_27685359190282
// MI455X (gfx1250) — compile-verified
//
#include <hip/hip_runtime.h>
#include <stdint.h>

typedef __attribute__((ext_vector_type(2))) float v2f;
typedef __attribute__((ext_vector_type(8))) float v8f;
typedef __attribute__((ext_vector_type(4))) int   v4i;

#define N_NODES 50000
#define N_EDGES 600000
#define FDIM    128
#define GDIM    64
#define MROWS   64            // rows per block in the node GEMM
#define LDS_STRIDE 132        // 128 + 4 pad -> conflict-free ds_load_b64

#if defined(__has_builtin)
#  if __has_builtin(__builtin_amdgcn_global_load_async_to_lds_b128)
#    define HAVE_ASYNC_LDS 1
#  endif
#  if __has_builtin(__builtin_amdgcn_s_wait_asynccnt)
#    define HAVE_WAIT_ASYNC 1
#  endif
#endif

#ifdef HAVE_ASYNC_LDS
typedef __attribute__((address_space(1))) v4i GV4;   // global v4i
typedef __attribute__((address_space(3))) v4i LV4;   // LDS v4i
#endif

// ------------------------------------------------------------------
// Transpose a 128x128 weight (out,in) -> (in,out) so WMMA B-tiles load
// coalesced (lane index runs along the output-column dimension).
// ------------------------------------------------------------------
__global__ __launch_bounds__(256) void transpose128(const float* __restrict__ W,
                                                    float* __restrict__ Wt) {
  int idx = blockIdx.x * blockDim.x + threadIdx.x;   // 0..16383
  int o = idx >> 7, k = idx & 127;
  Wt[k * 128 + o] = W[o * 128 + k];
}

// ------------------------------------------------------------------
// Fused 4-matrix node GEMM over a 64-row slab:
//   K = X Wk^T + bk, Q = X Wq^T + bq, V = X Wv^T + bv, H = X Ws^T + bs
// 256 threads = 8 waves; wave w owns output column tile [16w,16w+16)
// and iterates 4 row tiles, reusing each B-tile 4x (16 WMMA / k-step).
// X slab staged into LDS via GLOBAL_LOAD_ASYNC_TO_LDS_B128 when available.
// ------------------------------------------------------------------
__global__ __launch_bounds__(256)
void node_linear4(const float* __restrict__ X,
                  const float* __restrict__ WtK, const float* __restrict__ WtQ,
                  const float* __restrict__ WtV, const float* __restrict__ WtS,
                  const float* __restrict__ bK, const float* __restrict__ bQ,
                  const float* __restrict__ bV, const float* __restrict__ bS,
                  float* __restrict__ Ko, float* __restrict__ Qo,
                  float* __restrict__ Vo, float* __restrict__ Ho) {
  __shared__ float sA[MROWS * LDS_STRIDE];     // 33 KB
  const int r0 = blockIdx.x * MROWS;
  const int t  = threadIdx.x;

  // ---- stage: 64 rows x 128 floats, 8 x b128 chunks per thread ----
  {
#pragma unroll
    for (int i = 0; i < 8; ++i) {
      const int c    = i * 256 + t;        // 0..2047 16-byte chunks
      const int row  = c >> 5;             // 0..63
      const int col0 = (c & 31) * 4;       // 0..124
      int rr = r0 + row; if (rr >= N_NODES) rr = N_NODES - 1;  // clamp tail
      const float* gp = X + (size_t)rr * 128 + col0;
      float*       lp = &sA[row * LDS_STRIDE + col0];          // 16B aligned
#ifdef HAVE_ASYNC_LDS
      // AS(1) value == flat global address; AS(3) value == low 32 bits of flat
      __builtin_amdgcn_global_load_async_to_lds_b128(
          (GV4*)(uintptr_t)gp, (LV4*)(uint32_t)(uintptr_t)lp, 0, 0);
#else
      *reinterpret_cast<float4*>(lp) = *reinterpret_cast<const float4*>(gp);
#endif
    }
    if (r0 + MROWS < N_NODES)              // speculative prefetch of next slab
      __builtin_prefetch(X + (size_t)(r0 + MROWS) * 128 + t * 8, 0, 0);
#ifdef HAVE_ASYNC_LDS
#  ifdef HAVE_WAIT_ASYNC
    __builtin_amdgcn_s_wait_asynccnt(0);
#  else
    asm volatile("s_wait_asynccnt 0" ::: "memory");
#  endif
#endif
  }
  __syncthreads();

  const int lane = t & 31;
  const int wv   = t >> 5;        // 0..7 -> column tile
  const int lr   = lane & 15;
  const int half = lane >> 4;     // 0 or 1
  const int col  = wv * 16 + lr;

  v8f aK[4], aQ[4], aV[4], aS[4];
#pragma unroll
  for (int rt = 0; rt < 4; ++rt) { aK[rt] = (v8f){}; aQ[rt] = (v8f){}; aV[rt] = (v8f){}; aS[rt] = (v8f){}; }

  for (int kk = 0; kk < 32; ++kk) {
    const int kb = kk * 4;
    // B 4x16 fp32 (Wt is [k][o]): VGPR0 -> K = kb+2*half ; VGPR1 -> K = kb+1+2*half
    const int rB0 = (kb + 2 * half) * 128 + col;
    v2f bk, bq, bv_, bs;
    bk.x  = WtK[rB0];  bk.y  = WtK[rB0 + 128];
    bq.x  = WtQ[rB0];  bq.y  = WtQ[rB0 + 128];
    bv_.x = WtV[rB0];  bv_.y = WtV[rB0 + 128];
    bs.x  = WtS[rB0];  bs.y  = WtS[rB0 + 128];

#pragma unroll
    for (int rt = 0; rt < 4; ++rt) {
      // A 16x4 fp32: lanes 0-15 hold K=0,1 ; lanes 16-31 hold K=2,3
      const float* ap = &sA[(rt * 16 + lr) * LDS_STRIDE + kb + 2 * half];
      v2f a;  a.x = ap[0];  a.y = ap[1];
      aK[rt] = __builtin_amdgcn_wmma_f32_16x16x4_f32(false, a, false, bk,  (short)0, aK[rt], false, false);
      aQ[rt] = __builtin_amdgcn_wmma_f32_16x16x4_f32(false, a, false, bq,  (short)0, aQ[rt], false, false);
      aV[rt] = __builtin_amdgcn_wmma_f32_16x16x4_f32(false, a, false, bv_, (short)0, aV[rt], false, false);
      aS[rt] = __builtin_amdgcn_wmma_f32_16x16x4_f32(false, a, false, bs,  (short)0, aS[rt], false, false);
    }
  }

  const float bkc = bK[col], bqc = bQ[col], bvc = bV[col], bsc = bS[col];
  // C/D 16x16 fp32: VGPR g -> M = g (lanes 0-15) / M = g+8 (lanes 16-31)
#pragma unroll
  for (int rt = 0; rt < 4; ++rt) {
#pragma unroll
    for (int g = 0; g < 8; ++g) {
      const int r = r0 + rt * 16 + g + 8 * half;
      if (r < N_NODES) {
        Ko[(size_t)r * 128 + col] = aK[rt][g] + bkc;
        Qo[(size_t)r * 128 + col] = aQ[rt][g] + bqc;
        Vo[(size_t)r * 128 + col] = aV[rt][g] + bvc;
        Ho[(size_t)r * 128 + col] = aS[rt][g] + bsc;
      }
    }
  }
}

// ------------------------------------------------------------------
// Edge phase: one wave32 per edge; lane owns 4 features (float4).
// H[dst] += sigmoid(K[dst] + Q[src]) * V[src]   (fp32 global atomics,
// all per-node arrays are L2-resident: 4 x 25.6MB << 192MB L2).
// ------------------------------------------------------------------
__global__ __launch_bounds__(256)
void edge_scatter(const int* __restrict__ ei, const float* __restrict__ K,
                  const float* __restrict__ Q, const float* __restrict__ V,
                  float* __restrict__ H) {
  const int gid = blockIdx.x * blockDim.x + threadIdx.x;
  const int e = gid >> 5;
  if (e >= N_EDGES) return;
  const int lane = gid & 31;
  const int src = ei[e];
  const int dst = ei[N_EDGES + e];
  const int f0 = lane * 4;

  const float4 kv = *reinterpret_cast<const float4*>(K + (size_t)dst * 128 + f0);
  const float4 qv = *reinterpret_cast<const float4*>(Q + (size_t)src * 128 + f0);
  const float4 vv = *reinterpret_cast<const float4*>(V + (size_t)src * 128 + f0);

  float4 m;
  m.x = vv.x / (1.f + __expf(-(kv.x + qv.x)));
  m.y = vv.y / (1.f + __expf(-(kv.y + qv.y)));
  m.z = vv.z / (1.f + __expf(-(kv.z + qv.z)));
  m.w = vv.w / (1.f + __expf(-(kv.w + qv.w)));

  float* hp = H + (size_t)dst * 128 + f0;
  atomicAdd(hp + 0, m.x);
  atomicAdd(hp + 1, m.y);
  atomicAdd(hp + 2, m.z);
  atomicAdd(hp + 3, m.w);
}

__global__ __launch_bounds__(256) void relu_inplace(float* __restrict__ h, int n) {
  int i = blockIdx.x * blockDim.x + threadIdx.x;
  if (i < n) { float v = h[i]; h[i] = v > 0.f ? v : 0.f; }
}

// ------------------------------------------------------------------
// Pooling: float max via monotonic-uint encoding + atomicMax; sum + count.
// ------------------------------------------------------------------
__device__ __forceinline__ unsigned fenc(float x) {
  unsigned b = __float_as_uint(x);
  return (b & 0x80000000u) ? ~b : (b | 0x80000000u);
}
__device__ __forceinline__ float fdec(unsigned e) {
  return __uint_as_float((e & 0x80000000u) ? (e & 0x7FFFFFFFu) : ~e);
}

__global__ __launch_bounds__(256)
void pool_init(unsigned* __restrict__ gmax, float* __restrict__ gsum,
               float* __restrict__ cnt) {
  int i = blockIdx.x * blockDim.x + threadIdx.x;
  if (i < GDIM * FDIM) {
    gmax[i] = fenc(-__builtin_inff());
    gsum[i] = 0.f;
  }
  if (i < GDIM) cnt[i] = 0.f;
}

__global__ __launch_bounds__(128)
void pool(const float* __restrict__ H, const int* __restrict__ batch,
          unsigned* __restrict__ gmax, float* __restrict__ gsum,
          float* __restrict__ cnt) {
  const int n = blockIdx.x;
  const int f = threadIdx.x;
  const int g = batch[n];
  const float v = H[(size_t)n * 128 + f];
  atomicMax(&gmax[g * 128 + f], fenc(v));
  atomicAdd(&gsum[g * 128 + f], v);
  if (f == 0) atomicAdd(&cnt[g], 1.f);
}

// ------------------------------------------------------------------
// MLP head, single block (G=64 rows): concat(max,mean) -> L1+BN+ReLU ->
// L2+BN+ReLU -> L3.  All staging in LDS (96KB < 320KB/WG).
// ------------------------------------------------------------------
__global__ __launch_bounds__(256)
void head(const unsigned* __restrict__ gmax, const float* __restrict__ gsum,
          const float* __restrict__ cnt,
          const float* __restrict__ W1, const float* __restrict__ b1,
          const float* __restrict__ g1, const float* __restrict__ be1,
          const float* __restrict__ W2, const float* __restrict__ b2,
          const float* __restrict__ g2, const float* __restrict__ be2,
          const float* __restrict__ W3, const float* __restrict__ b3,
          float* __restrict__ out) {
  __shared__ float sf[GDIM * 256];   // pooled features; reused as sh2 in stage C
  __shared__ float sh1[GDIM * 128];
  const int t = threadIdx.x;

  for (int idx = t; idx < GDIM * 256; idx += 256) {
    const int g = idx >> 8, j = idx & 255;
    float val;
    if (j < 128) {
      val = fdec(gmax[g * 128 + j]);
    } else {
      float c = cnt[g]; c = c < 1.f ? 1.f : c;
      val = gsum[g * 128 + (j - 128)] / c;
    }
    sf[g * 256 + j] = val;
  }
  __syncthreads();

  if (t < 128) {  // layer 1: column t
    float s = 0.f, ss = 0.f;
    for (int g = 0; g < GDIM; ++g) {
      float acc = b1[t];
      for (int k = 0; k < 256; ++k) acc += sf[g * 256 + k] * W1[t * 256 + k];
      sh1[g * 128 + t] = acc; s += acc; ss += acc * acc;
    }
    const float m = s * (1.f / GDIM);
    const float var = ss * (1.f / GDIM) - m * m;
    const float is = rsqrtf(var + 1e-5f);
    const float ga = g1[t], bb = be1[t];
    for (int g = 0; g < GDIM; ++g) {
      float h = ga * (sh1[g * 128 + t] - m) * is + bb;
      sh1[g * 128 + t] = h > 0.f ? h : 0.f;
    }
  }
  __syncthreads();

  float* sh2 = sf;  // sf is dead now
  if (t < 64) {     // layer 2: column t
    float s = 0.f, ss = 0.f;
    for (int g = 0; g < GDIM; ++g) {
      float acc = b2[t];
      for (int k = 0; k < 128; ++k) acc += sh1[g * 128 + k] * W2[t * 128 + k];
      sh2[g * 64 + t] = acc; s += acc; ss += acc * acc;
    }
    const float m = s * (1.f / GDIM);
    const float var = ss * (1.f / GDIM) - m * m;
    const float is = rsqrtf(var + 1e-5f);
    const float ga = g2[t], bb = be2[t];
    for (int g = 0; g < GDIM; ++g) {
      float h = ga * (sh2[g * 64 + t] - m) * is + bb;
      sh2[g * 64 + t] = h > 0.f ? h : 0.f;
    }
  }
  __syncthreads();

  if (t < 64) {     // layer 3: one graph per thread
    float acc = b3[0];
    for (int k = 0; k < 64; ++k) acc += sh2[t * 64 + k] * W3[k];
    out[t] = acc;
  }
}

// ------------------------------------------------------------------
extern "C" void kernel_launch(void* const* d_in, const int* in_sizes, int n_in,
                              void* d_out, int out_size, void* d_ws, size_t ws_size,
                              hipStream_t stream) {
  (void)in_sizes; (void)n_in; (void)out_size; (void)ws_size;

  const float* x     = (const float*)d_in[0];
  const int*   ei    = (const int*)d_in[1];
  const int*   batch = (const int*)d_in[2];
  // layer weights/biases (dict order): l{0,1}_{Wk,bk,Wq,bq,Wv,bv,Ws,bs}
  const float* Wmat[8]; const float* bvec[8];
  for (int i = 0; i < 8; ++i) {
    Wmat[i] = (const float*)d_in[3 + 2 * i];
    bvec[i] = (const float*)d_in[4 + 2 * i];
  }
  const float* W1  = (const float*)d_in[19];
  const float* b1  = (const float*)d_in[20];
  const float* g1  = (const float*)d_in[21];
  const float* be1 = (const float*)d_in[22];
  const float* W2  = (const float*)d_in[23];
  const float* b2  = (const float*)d_in[24];
  const float* g2  = (const float*)d_in[25];
  const float* be2 = (const float*)d_in[26];
  const float* W3  = (const float*)d_in[27];
  const float* b3  = (const float*)d_in[28];

  // workspace carve-up
  float* Wt   = (float*)d_ws;                       // 8 * 128*128
  float* K    = Wt + 8 * 16384;
  float* Q    = K  + (size_t)N_NODES * 128;
  float* V    = Q  + (size_t)N_NODES * 128;
  float* H0   = V  + (size_t)N_NODES * 128;
  float* H1   = H0 + (size_t)N_NODES * 128;
  unsigned* gmax = (unsigned*)(H1 + (size_t)N_NODES * 128);
  float* gsum = (float*)(gmax + GDIM * 128);
  float* cnt  = gsum + GDIM * 128;

  for (int i = 0; i < 8; ++i)
    transpose128<<<64, 256, 0, stream>>>(Wmat[i], Wt + (size_t)i * 16384);

  const int rowBlocks = (N_NODES + MROWS - 1) / MROWS;   // 782
  // ---- layer 0 ----
  node_linear4<<<rowBlocks, 256, 0, stream>>>(
      x, Wt + 0 * 16384, Wt + 1 * 16384, Wt + 2 * 16384, Wt + 3 * 16384,
      bvec[0], bvec[1], bvec[2], bvec[3], K, Q, V, H0);
  edge_scatter<<<(N_EDGES * 32) / 256, 256, 0, stream>>>(ei, K, Q, V, H0);
  relu_inplace<<<(N_NODES * 128) / 256, 256, 0, stream>>>(H0, N_NODES * 128);
  // ---- layer 1 ----
  node_linear4<<<rowBlocks, 256, 0, stream>>>(
      H0, Wt + 4 * 16384, Wt + 5 * 16384, Wt + 6 * 16384, Wt + 7 * 16384,
      bvec[4], bvec[5], bvec[6], bvec[7], K, Q, V, H1);
  edge_scatter<<<(N_EDGES * 32) / 256, 256, 0, stream>>>(ei, K, Q, V, H1);
  // ---- pooling ----
  pool_init<<<(GDIM * FDIM + 255) / 256, 256, 0, stream>>>(gmax, gsum, cnt);
  pool<<<N_NODES, 128, 0, stream>>>(H1, batch, gmax, gsum, cnt);
  // ---- head ----
  head<<<1, 256, 0, stream>>>(gmax, gsum, cnt, W1, b1, g1, be1,
                              W2, b2, g2, be2, W3, b3, (float*)d_out);
}